// LightGCNModule_72000831750473
// MI455X (gfx1250) — compile-verified
//
#include <hip/hip_runtime.h>
#include <hip/hip_bf16.h>

#define N_USERS   100000
#define N_ITEMS   50000
#define N_TOTAL   150000   // N_USERS + N_ITEMS
#define LATENT    64
#define BATCH     4096
#define N_EDGES   1200000  // 2 * N_INTERACTIONS

typedef __attribute__((ext_vector_type(2))) float v2f;
typedef __attribute__((ext_vector_type(8))) float v8f;

// ---------------------------------------------------------------------------
// Zero a 150000x64 float buffer (2.4M float4, exact grid: 9375 x 256).
// ---------------------------------------------------------------------------
__global__ void lgcn_zero(float4* __restrict__ p) {
    unsigned i = blockIdx.x * blockDim.x + threadIdx.x;
    p[i] = make_float4(0.f, 0.f, 0.f, 0.f);
}

// ---------------------------------------------------------------------------
// SPMM via scatter-atomics: one wave (32 lanes) per edge, float2 per lane.
// Eout[row[e], :] += vals[e] * Ein[col[e], :]   (L2-resident, 38.4MB << 192MB)
// ---------------------------------------------------------------------------
__global__ void lgcn_spmm(const int*   __restrict__ row,
                          const int*   __restrict__ col,
                          const float* __restrict__ vals,
                          const float* __restrict__ Ein,
                          float*       __restrict__ Eout) {
    unsigned gid = blockIdx.x * blockDim.x + threadIdx.x;
    unsigned e   = gid >> 5;
    if (e >= N_EDGES) return;
    unsigned t = gid & 31;

    int   r = row[e];
    int   c = col[e];
    float v = vals[e];

    float2 x = ((const float2*)(Ein + (size_t)c * LATENT))[t];
    float* dst = Eout + (size_t)r * LATENT + (size_t)t * 2;

    __hip_atomic_fetch_add(dst,     v * x.x, __ATOMIC_RELAXED, __HIP_MEMORY_SCOPE_AGENT);
    __hip_atomic_fetch_add(dst + 1, v * x.y, __ATOMIC_RELAXED, __HIP_MEMORY_SCOPE_AGENT);
}

// ---------------------------------------------------------------------------
// Initialize all 6 output segments from E0 (acc starts at E0; segs 3..5 are
// the *_emb0 outputs).  tid covers 6*4096*64 = 1,572,864 (6144 x 256 exact).
// ---------------------------------------------------------------------------
__global__ void lgcn_init_out(float*       __restrict__ out,
                              const int*   __restrict__ users,
                              const int*   __restrict__ pos,
                              const int*   __restrict__ neg,
                              const float* __restrict__ E0) {
    unsigned tid = blockIdx.x * blockDim.x + threadIdx.x;
    unsigned d   = tid & 63;
    unsigned i   = (tid >> 6) & (BATCH - 1);
    unsigned seg = tid >> 18;                 // 4096*64 = 2^18
    unsigned s3  = (seg >= 3) ? seg - 3 : seg;
    int ridx = (s3 == 0) ? users[i]
             : (s3 == 1) ? (N_USERS + pos[i])
                         : (N_USERS + neg[i]);
    out[tid] = E0[(size_t)ridx * LATENT + d];
}

// ---------------------------------------------------------------------------
// Accumulate one layer's embedding at the gathered rows into segments 0..2.
// tid covers 3*4096*64 = 786,432 (3072 x 256 exact).
// ---------------------------------------------------------------------------
__global__ void lgcn_accum(float*       __restrict__ out,
                           const int*   __restrict__ users,
                           const int*   __restrict__ pos,
                           const int*   __restrict__ neg,
                           const float* __restrict__ Ek) {
    unsigned tid = blockIdx.x * blockDim.x + threadIdx.x;
    unsigned d   = tid & 63;
    unsigned i   = (tid >> 6) & (BATCH - 1);
    unsigned seg = tid >> 18;
    int ridx = (seg == 0) ? users[i]
             : (seg == 1) ? (N_USERS + pos[i])
                          : (N_USERS + neg[i]);
    out[tid] += Ek[(size_t)ridx * LATENT + d];
}

// ---------------------------------------------------------------------------
// Final scale by 1/(N_LAYERS+1) = 0.25 on segments 0..2 via WMMA:
//   D = (0.25 * I16) * X,  K=16 accumulated as 4 chained 16x16x4 f32 WMMAs.
// Exact in fp32 (multiply by 0.25 / 0 / 1 is exact).  One wave per 256-float
// tile; 3072 tiles total (384 blocks x 8 waves, exact -> EXEC all ones).
// ---------------------------------------------------------------------------
__global__ void lgcn_scale_wmma(float* __restrict__ out) {
    unsigned wave = (blockIdx.x * blockDim.x + threadIdx.x) >> 5;
    unsigned lane = threadIdx.x & 31;
    float* tile = out + (size_t)wave * 256;

    int n = lane & 15;        // column for B/C/D; also M index for A
    int h = lane >> 4;        // half-wave selector

    v8f d = {0.f, 0.f, 0.f, 0.f, 0.f, 0.f, 0.f, 0.f};
#pragma unroll
    for (int j = 0; j < 4; ++j) {
        int k0 = 4 * j + 2 * h;           // this lane's K pair within chunk j
        v2f a, b;
        // A_j[m,k] = 0.25 if m == 4j+k else 0  (identity slice, scaled)
        a.x = (n == k0)     ? 0.25f : 0.0f;
        a.y = (n == k0 + 1) ? 0.25f : 0.0f;
        // B_j[k,n] = X[4j+k, n]  (tile stored row-major, 16x16)
        b.x = tile[k0 * 16 + n];
        b.y = tile[(k0 + 1) * 16 + n];
        d = __builtin_amdgcn_wmma_f32_16x16x4_f32(
                /*neg_a=*/false, a, /*neg_b=*/false, b,
                /*c_mod=*/(short)0, d, /*reuse_a=*/false, /*reuse_b=*/false);
    }
#pragma unroll
    for (int v = 0; v < 8; ++v)
        tile[(v + 8 * h) * 16 + n] = d[v];   // D row = vgpr + 8h, col = n
}

// ---------------------------------------------------------------------------
extern "C" void kernel_launch(void* const* d_in, const int* in_sizes, int n_in,
                              void* d_out, int out_size, void* d_ws, size_t ws_size,
                              hipStream_t stream) {
    (void)in_sizes; (void)n_in; (void)out_size; (void)ws_size;

    const int*   users = (const int*)  d_in[0];
    const int*   pos   = (const int*)  d_in[1];
    const int*   neg   = (const int*)  d_in[2];
    const float* E0    = (const float*)d_in[3];
    const int*   arow  = (const int*)  d_in[4];
    const int*   acol  = (const int*)  d_in[5];
    const float* avals = (const float*)d_in[6];
    float* out = (float*)d_out;

    // Two ping-pong layer buffers: 2 * 150000*64*4B = 76.8 MB of workspace.
    float* bufA = (float*)d_ws;
    float* bufB = bufA + (size_t)N_TOTAL * LATENT;

    dim3 blk(256);

    // acc := E0 at gathered rows; also fills the *_emb0 output segments.
    lgcn_init_out<<<6144, blk, 0, stream>>>(out, users, pos, neg, E0);

    // Layer 1: E1 = S*E0 -> bufA ; acc += E1[rows]
    lgcn_zero <<<9375,   blk, 0, stream>>>((float4*)bufA);
    lgcn_spmm <<<150000, blk, 0, stream>>>(arow, acol, avals, E0, bufA);
    lgcn_accum<<<3072,   blk, 0, stream>>>(out, users, pos, neg, bufA);

    // Layer 2: E2 = S*E1 -> bufB ; acc += E2[rows]
    lgcn_zero <<<9375,   blk, 0, stream>>>((float4*)bufB);
    lgcn_spmm <<<150000, blk, 0, stream>>>(arow, acol, avals, bufA, bufB);
    lgcn_accum<<<3072,   blk, 0, stream>>>(out, users, pos, neg, bufB);

    // Layer 3: E3 = S*E2 -> bufA ; acc += E3[rows]
    lgcn_zero <<<9375,   blk, 0, stream>>>((float4*)bufA);
    lgcn_spmm <<<150000, blk, 0, stream>>>(arow, acol, avals, bufB, bufA);
    lgcn_accum<<<3072,   blk, 0, stream>>>(out, users, pos, neg, bufA);

    // final_emb = acc / 4 on segments 0..2, via v_wmma_f32_16x16x4_f32.
    lgcn_scale_wmma<<<384, blk, 0, stream>>>(out);
}